// Binarize_52913997086767
// MI455X (gfx1250) — compile-verified
//
#include <hip/hip_runtime.h>
#include <stdint.h>

// Binarize + big-endian bitpack, CDNA5 (gfx1250) version.
//
// x: [B=2048, N=32768] f32, ths: [3] f32  ->  out: uint8 [B, 3, N/8]
// Written as uint32 words (N/32 = 1024 words per row per threshold).
//
// Bandwidth-bound: 256MB read + 24MB write ~= 12us @ 23.3 TB/s.
// Data path: async global->LDS b128 copies (ASYNCcnt, double buffered),
// packing via wave32 ballot + s_brev/bswap (32 predicate bits -> 1 word).

#define TPB            256
#define N_FIXED        32768
#define STAGE_FLOATS   1024                       // 4KB: one b128 async per thread
#define NSTAGES        (N_FIXED / STAGE_FLOATS)   // 32
#define WORDS_PER_ROW  (N_FIXED / 32)             // 1024
#define WORDS_PER_STG  (STAGE_FLOATS / 32)        // 32
#define GROUPS_PER_WV  (WORDS_PER_STG / (TPB/32)) // 4

__device__ __forceinline__ uint32_t ballot32(bool p) {
#if __has_builtin(__builtin_amdgcn_ballot_w32)
  return __builtin_amdgcn_ballot_w32(p);
#else
  return (uint32_t)__ballot(p);   // wave32: low 32 bits
#endif
}

// ballot bit i = pred(element i). packbits is big-endian per byte:
// byte k, bit (7-j) = pred(element 8k+j). bswap(brev(m)) realizes exactly that.
__device__ __forceinline__ uint32_t pack_be(uint32_t m) {
  return __builtin_bswap32(__builtin_bitreverse32(m));
}

__global__ __launch_bounds__(TPB) void binarize_pack_kernel(
    const float* __restrict__ x, const float* __restrict__ ths,
    uint32_t* __restrict__ out) {
  __shared__ float buf[2][STAGE_FLOATS];

  const int b    = blockIdx.x;
  const int tid  = threadIdx.x;
  const int lane = tid & 31;
  const int wave = tid >> 5;

  const float th0 = ths[0];
  const float th1 = ths[1];
  const float th2 = ths[2];

  const float* rowp = x + (size_t)b * N_FIXED;

  // Issue one 16B async copy per thread: stage s, buffer bi.
  auto issue = [&](int s, int bi) {
    const float* gp = rowp + s * STAGE_FLOATS + tid * 4;
    uint32_t lds_addr = (uint32_t)(uintptr_t)(&buf[bi][tid * 4]);
    asm volatile("global_load_async_to_lds_b128 %0, %1, off"
                 :: "v"(lds_addr), "v"(gp)
                 : "memory");
  };

  issue(0, 0);

  for (int s = 0; s < NSTAGES; ++s) {
    const int bi = s & 1;

    if (s + 1 < NSTAGES) {
      // Prefetch next stage into the other buffer; its previous consumers
      // finished at the trailing barrier of iteration s-1.
      issue(s + 1, bi ^ 1);
      // Async copies complete in order: cnt<=1 means stage s has landed.
      asm volatile("s_wait_asynccnt 0x1" ::: "memory");
    } else {
      asm volatile("s_wait_asynccnt 0x0" ::: "memory");
    }
    __syncthreads();  // all waves' chunks of stage s visible in LDS

    uint32_t* orow = out + (size_t)b * 3u * WORDS_PER_ROW + (uint32_t)(s * WORDS_PER_STG);

#pragma unroll
    for (int j = 0; j < GROUPS_PER_WV; ++j) {
      const int g = wave * GROUPS_PER_WV + j;
      const float v = buf[bi][g * 32 + lane];

      const uint32_t w0 = pack_be(ballot32(v > th0));
      const uint32_t w1 = pack_be(ballot32(v > th1));
      const uint32_t w2 = pack_be(ballot32(v > th2));

      if (lane == 0) {
        orow[g]                     = w0;
        orow[g + WORDS_PER_ROW]     = w1;
        orow[g + 2 * WORDS_PER_ROW] = w2;
      }
    }

    __syncthreads();  // reads of buf[bi] done before it is refilled at s+1
  }
}

extern "C" void kernel_launch(void* const* d_in, const int* in_sizes, int n_in,
                              void* d_out, int out_size, void* d_ws, size_t ws_size,
                              hipStream_t stream) {
  (void)n_in; (void)out_size; (void)d_ws; (void)ws_size;

  const float* x   = (const float*)d_in[0];   // [B, N] f32
  const float* ths = (const float*)d_in[1];   // [3] f32

  uint32_t* out = (uint32_t*)d_out;           // uint8 [B,3,N/8] viewed as u32 words

  const int B = in_sizes[0] / N_FIXED;        // 2048

  binarize_pack_kernel<<<B, TPB, 0, stream>>>(x, ths, out);
}